// NoiseReduction_36618891165819
// MI455X (gfx1250) — compile-verified
//
#include <hip/hip_runtime.h>

// ---------------------------------------------------------------------------
// Savitzky-Golay (W=11, poly=3) over (64,16,65536) f32.
// Bandwidth-bound (AI ~2.9 FLOP/B, ~22us floor at 23.3 TB/s): stream via
// CDNA5 async global->LDS (ASYNCcnt double-buffered), compute interior FIR
// as banded-Toeplitz matmul on v_wmma_f32_16x16x4_f32.
// ---------------------------------------------------------------------------

typedef float v2f __attribute__((ext_vector_type(2)));
typedef float v8f __attribute__((ext_vector_type(8)));

#define WINDOW 11
#define HALF   5

#if __has_builtin(__builtin_amdgcn_wmma_f32_16x16x4_f32)
#define HAVE_WMMA4 1
#else
#define HAVE_WMMA4 0
#endif

#if __has_builtin(__builtin_amdgcn_global_load_async_to_lds_b32)
#define HAVE_ASYNC 1
#else
#define HAVE_ASYNC 0
#endif

// Classic SG smoothing taps for window 11, degree 2/3 (== pinv row 0).
__constant__ float C_TAPS[WINDOW] = {
    -36.f/429.f,  9.f/429.f, 44.f/429.f, 69.f/429.f, 84.f/429.f, 89.f/429.f,
     84.f/429.f, 69.f/429.f, 44.f/429.f,  9.f/429.f, -36.f/429.f };

__device__ __forceinline__ float tap(int t) {
    return (t >= 0 && t < WINDOW) ? C_TAPS[t] : 0.0f;
}

__device__ __forceinline__ void async_copy_f32(const float* g, float* l) {
#if HAVE_ASYNC
    __builtin_amdgcn_global_load_async_to_lds_b32(
        (__attribute__((address_space(1))) int*)(void*)g,
        (__attribute__((address_space(3))) int*)(void*)l,
        0, 0);
#else
    *l = *g;
#endif
}

__device__ __forceinline__ void wait_async0() {
#if HAVE_ASYNC
#if __has_builtin(__builtin_amdgcn_s_wait_asynccnt)
    __builtin_amdgcn_s_wait_asynccnt(0);
#else
    asm volatile("s_wait_asynccnt 0" ::: "memory");
#endif
#endif
}

#define BLOCK_T      128           // 4 waves (wave32)
#define WAVE_OUT     256           // outputs per wave per tile (16x16 D)
#define WG_OUT       1024          // outputs per WG per tile
#define TILES_PER_WG 8
#define WGS_PER_ROW  8             // 8*8*1024 = 65536 >= 65526 interior outs
#define BUF_N        1040          // WG input window: WG_OUT + halo
#define BUF_STRIDE   1056
#define NLOAD        9             // ceil(1040/128)

__global__ __launch_bounds__(BLOCK_T)
void sg_interior_kernel(const float* __restrict__ x,
                        float* __restrict__ out, int L) {
    __shared__ float lds[2 * BUF_STRIDE];

    const int row  = blockIdx.y;
    const float* xr = x   + (size_t)row * (size_t)L;
    float*       yr = out + (size_t)row * (size_t)L;

    const int tid  = threadIdx.x;
    const int lane = tid & 31;
    const int wv   = tid >> 5;
    const int m    = lane & 15;   // A-row M / D-col N / B-col n
    const int hh   = lane >> 4;   // lane half: selects K pair / D row half

#if HAVE_WMMA4
    // Toeplitz coefficient matrices B_b[k'][n] = c[k' + 16b - n], packed in
    // the B operand layout: lane -> (K = 4kb + 2*hh + {0,1}, N = m).
    // (b=1,kb=3) is structurally zero (t >= 13 > 10) and is skipped below.
    v2f bc[7];
#pragma unroll
    for (int b = 0; b < 2; ++b)
#pragma unroll
        for (int kb = 0; kb < 4; ++kb) {
            if (b == 1 && kb == 3) continue;
            const int K0 = 4 * kb + 2 * hh + 16 * b;
            v2f v;
            v.x = tap(K0     - m);
            v.y = tap(K0 + 1 - m);
            bc[b * 4 + kb] = v;
        }
#endif

    const int base_j0 = HALF + blockIdx.x * (WG_OUT * TILES_PER_WG);

    // Prefetch tile 0 (async global -> LDS, ASYNCcnt-tracked).
    {
        const int gs = base_j0 - HALF;
        float* dst = &lds[0];
#pragma unroll
        for (int r2 = 0; r2 < NLOAD; ++r2) {
            int i  = r2 * BLOCK_T + tid;
            int ii = i < BUF_N ? i : (BUF_N - 1);   // uniform ASYNCcnt/lane
            int gi = gs + ii; gi = gi < L ? gi : (L - 1);  // row-end clamp
            async_copy_f32(xr + gi, dst + ii);
        }
    }

    for (int t = 0; t < TILES_PER_WG; ++t) {
        wait_async0();        // my tile-t loads landed in LDS
        __syncthreads();      // everyone's landed; prev-tile readers done
        if (t + 1 < TILES_PER_WG) {   // prefetch next tile into other buffer
            const int gs = base_j0 - HALF + (t + 1) * WG_OUT;
            float* dst = &lds[((t + 1) & 1) * BUF_STRIDE];
#pragma unroll
            for (int r2 = 0; r2 < NLOAD; ++r2) {
                int i  = r2 * BLOCK_T + tid;
                int ii = i < BUF_N ? i : (BUF_N - 1);
                int gi = gs + ii; gi = gi < L ? gi : (L - 1);
                async_copy_f32(xr + gi, dst + ii);
            }
        }

        const float* wb = &lds[(t & 1) * BUF_STRIDE + wv * WAVE_OUT];
        const int j0 = base_j0 + t * WG_OUT + wv * WAVE_OUT;

#if HAVE_WMMA4
        // D = A0*B0 + A1*B1 as 7 chained 16x16x4 f32 WMMAs
        // (the 8th K-chunk has an all-zero coefficient band).
        v8f acc = {0.f, 0.f, 0.f, 0.f, 0.f, 0.f, 0.f, 0.f};
#pragma unroll
        for (int b = 0; b < 2; ++b)
#pragma unroll
            for (int kb = 0; kb < 4; ++kb) {
                if (b == 1 && kb == 3) continue;
                const int o = 16 * m + 16 * b + 4 * kb + 2 * hh;
                v2f a; a.x = wb[o]; a.y = wb[o + 1];   // ds_load_b64 pair
                acc = __builtin_amdgcn_wmma_f32_16x16x4_f32(
                    false, a, false, bc[b * 4 + kb],
                    (short)0, acc, false, false);
            }
        // D layout: VGPR r, lane -> (M = r + 8*hh, N = m); y[j0 + 16M + N].
        float* yb = yr + j0 + 128 * hh + m;            // lane base, +16 per r
        const bool full = (j0 + WAVE_OUT - 1) <= (L - (HALF + 1));
        if (full) {                                    // wave-uniform branch
#pragma unroll
            for (int r2 = 0; r2 < 8; ++r2)
                yb[16 * r2] = acc[r2];                 // const-offset stores
        } else {                                       // only last wave-tile
#pragma unroll
            for (int r2 = 0; r2 < 8; ++r2) {
                int j = j0 + 16 * (r2 + 8 * hh) + m;
                if (j <= L - (HALF + 1)) yr[j] = acc[r2];
            }
        }
#else
        // VALU fallback: 8 outputs/thread, 11-tap FIR from LDS.
#pragma unroll
        for (int r2 = 0; r2 < 8; ++r2) {
            int o = lane + 32 * r2;
            int j = j0 + o;
            float s = 0.f;
#pragma unroll
            for (int k = 0; k < WINDOW; ++k)
                s = fmaf(C_TAPS[k], wb[o + k], s);
            if (j <= L - (HALF + 1)) yr[j] = s;
        }
#endif
    }
}

// ---------------------------------------------------------------------------
// Edges: cubic LS fit of the first/last 11 samples, evaluated at 0..4.
// Solve 4x4 normal equations in f64 (Gram matrix = power sums of 0..10).
// ---------------------------------------------------------------------------
__global__ void sg_edge_kernel(const float* __restrict__ x,
                               float* __restrict__ out, int rows, int L) {
    int tid = blockIdx.x * blockDim.x + threadIdx.x;
    if (tid >= 2 * rows) return;
    int row  = tid >> 1;
    int side = tid & 1;                       // 0 = left, 1 = right
    const float* xr = x   + (size_t)row * (size_t)L;
    float*       yr = out + (size_t)row * (size_t)L;

    double rhs[4] = {0.0, 0.0, 0.0, 0.0};
    for (int i = 0; i < WINDOW; ++i) {
        double xi = (double)(side == 0 ? xr[i] : xr[L - 1 - i]);
        double p = 1.0;
        for (int j = 0; j < 4; ++j) { rhs[j] += p * xi; p *= (double)i; }
    }
    // M = V^T V, V = Vandermonde(0..10, deg 3): power sums S0..S6.
    double M[4][5] = {
        {  11.0,    55.0,    385.0,    3025.0, rhs[0]},
        {  55.0,   385.0,   3025.0,   25333.0, rhs[1]},
        { 385.0,  3025.0,  25333.0,  220825.0, rhs[2]},
        {3025.0, 25333.0, 220825.0, 1978405.0, rhs[3]}};
    for (int c = 0; c < 4; ++c) {            // Gauss w/ partial pivoting
        int piv = c; double best = fabs(M[c][c]);
        for (int r2 = c + 1; r2 < 4; ++r2) {
            double v = fabs(M[r2][c]);
            if (v > best) { best = v; piv = r2; }
        }
        if (piv != c)
            for (int k = c; k < 5; ++k) {
                double tmp = M[c][k]; M[c][k] = M[piv][k]; M[piv][k] = tmp;
            }
        double inv = 1.0 / M[c][c];
        for (int r2 = c + 1; r2 < 4; ++r2) {
            double f = M[r2][c] * inv;
            for (int k = c; k < 5; ++k) M[r2][k] -= f * M[c][k];
        }
    }
    double beta[4];
    for (int c = 3; c >= 0; --c) {
        double s = M[c][4];
        for (int k = c + 1; k < 4; ++k) s -= M[c][k] * beta[k];
        beta[c] = s / M[c][c];
    }
    for (int h = 0; h < HALF; ++h) {
        double hv = (double)h;
        double v = ((beta[3] * hv + beta[2]) * hv + beta[1]) * hv + beta[0];
        if (side == 0) yr[h] = (float)v;
        else           yr[L - 1 - h] = (float)v;
    }
}

// ---------------------------------------------------------------------------
extern "C" void kernel_launch(void* const* d_in, const int* in_sizes, int n_in,
                              void* d_out, int out_size, void* d_ws,
                              size_t ws_size, hipStream_t stream) {
    const float* x = (const float*)d_in[0];
    float* out = (float*)d_out;
    const int L = 65536;
    const int rows = in_sizes[0] / L;        // 64*16 = 1024

    dim3 gridI(WGS_PER_ROW, rows);
    sg_interior_kernel<<<gridI, BLOCK_T, 0, stream>>>(x, out, L);

    const int nEdge = rows * 2;
    const int eb = 128;
    sg_edge_kernel<<<(nEdge + eb - 1) / eb, eb, 0, stream>>>(x, out, rows, L);
}